// Encoder_Decoder_35897336660124
// MI455X (gfx1250) — compile-verified
//
#include <hip/hip_runtime.h>
#include <math.h>

// Problem constants (from setup_inputs): B=64, S=256, T=256, D=512, tau=0.3
#define B_ 64
#define S_ 256
#define T_ 256
#define D_ 512
#define TAU_ 0.3f
#define NEG_INF (-__builtin_inff())

typedef __attribute__((ext_vector_type(2))) float v2f;
typedef __attribute__((ext_vector_type(8))) float v8f;

// CDNA5 f32 WMMA: D(16x16,f32) = A(16x4,f32) * B(4x16,f32) + C
// 8-arg form: (neg_a, A, neg_b, B, c_mod, C, reuse_a, reuse_b)
__device__ __forceinline__ v8f wmma_f32_16x16x4(v2f a, v2f b, v8f c) {
  return __builtin_amdgcn_wmma_f32_16x16x4_f32(false, a, false, b, (short)0, c,
                                               false, false);
}

// ---------------------------------------------------------------------------
// Kernel 1: soft_copy. One block = (batch b, 16-row tile of T).
// Phase A: compute softmax weights W[16][256] into LDS.
// Phase B: W[16,256] x enc[256,512] via v_wmma_f32_16x16x4_f32.
// 8 waves, each wave owns 64 output columns (4 N-tiles of 16).
// ---------------------------------------------------------------------------
__global__ __launch_bounds__(256) void soft_copy_kernel(
    const float* __restrict__ enc,          // [B,S,D]
    const unsigned char* __restrict__ smask,// [B,S] bool bytes
    const unsigned char* __restrict__ tmask,// [B,T] bool bytes
    float* __restrict__ dec)                // [B,T,D]
{
  __shared__ float wld[16][S_ + 2];         // padded row stride (even, 8B align)
  __shared__ unsigned char sm[S_];
  const int tid = threadIdx.x;
  const int b = blockIdx.y;
  const int t0 = blockIdx.x * 16;

  sm[tid] = smask[b * S_ + tid];
  __syncthreads();

  // --- softmax of -|t-s|/tau over masked s; 16 threads per row ---
  {
    const int r = tid >> 4;
    const int q = tid & 15;
    const int t = t0 + r;
    const float tm = tmask[b * T_ + t] ? 1.0f : 0.0f;
    float mx = NEG_INF;
    for (int j = 0; j < 16; ++j) {
      int s = q * 16 + j;
      if (sm[s]) mx = fmaxf(mx, -fabsf((float)(t - s)) * (1.0f / TAU_));
    }
    for (int off = 8; off; off >>= 1)
      mx = fmaxf(mx, __shfl_xor(mx, off, 16));
    float sum = 0.f;
    for (int j = 0; j < 16; ++j) {
      int s = q * 16 + j;
      float e = 0.f;
      if (sm[s]) e = expf(-fabsf((float)(t - s)) * (1.0f / TAU_) - mx);
      wld[r][s] = e;
      sum += e;
    }
    for (int off = 8; off; off >>= 1)
      sum += __shfl_xor(sum, off, 16);
    const float inv = tm / sum;               // zero row if tgt_mask false
    for (int j = 0; j < 16; ++j) wld[r][q * 16 + j] *= inv;
  }
  __syncthreads();

  // --- WMMA GEMM: acc[16,64 per wave] += W[16,4] * enc[4,64] ---
  const int lane = tid & 31;
  const int wave = tid >> 5;
  const int row16 = lane & 15;
  const int khalf = (lane >> 4) << 1;         // 0 for lanes 0-15, 2 for 16-31
  const int nb = wave * 64;
  const float* encB = enc + (size_t)b * S_ * D_;

  v8f acc0 = {}, acc1 = {}, acc2 = {}, acc3 = {};
  for (int k0 = 0; k0 < S_; k0 += 4) {
    v2f a;                                    // A frag from LDS (broadcast)
    a.x = wld[row16][k0 + khalf];
    a.y = wld[row16][k0 + khalf + 1];
    const float* p0 = encB + (size_t)(k0 + khalf) * D_;
    const float* p1 = p0 + D_;
    v2f b0, b1, b2, b3;                       // B frags, 64B-coalesced halves
    b0.x = p0[nb + row16];       b0.y = p1[nb + row16];
    b1.x = p0[nb + 16 + row16];  b1.y = p1[nb + 16 + row16];
    b2.x = p0[nb + 32 + row16];  b2.y = p1[nb + 32 + row16];
    b3.x = p0[nb + 48 + row16];  b3.y = p1[nb + 48 + row16];
    acc0 = wmma_f32_16x16x4(a, b0, acc0);
    acc1 = wmma_f32_16x16x4(a, b1, acc1);
    acc2 = wmma_f32_16x16x4(a, b2, acc2);
    acc3 = wmma_f32_16x16x4(a, b3, acc3);
  }

  // D layout: VGPR j -> M=j (lanes 0-15) / M=j+8 (lanes 16-31), N=lane&15
  const int rhi = (lane >> 4) * 8;
  float* decB = dec + ((size_t)b * T_ + t0) * D_;
  for (int j = 0; j < 8; ++j) {
    const size_t ro = (size_t)(rhi + j) * D_;
    decB[ro + nb + row16]      = acc0[j];
    decB[ro + nb + 16 + row16] = acc1[j];
    decB[ro + nb + 32 + row16] = acc2[j];
    decB[ro + nb + 48 + row16] = acc3[j];
  }
}

// ---------------------------------------------------------------------------
// Kernel 2: row L2 norms of dec_input and tgt_emb (one wave per row).
// ---------------------------------------------------------------------------
__global__ __launch_bounds__(256) void norms_kernel(
    const float* __restrict__ dec,  // [B,T,D]
    const float* __restrict__ tgt,  // [B,T,D]
    float* __restrict__ dec_n,      // [B*T]
    float* __restrict__ tgt_n)      // [B*T]
{
  const int lane = threadIdx.x & 31;
  int row = blockIdx.x * 8 + (threadIdx.x >> 5);
  const float* src;
  float* dst;
  if (row < B_ * T_) { src = dec + (size_t)row * D_; dst = dec_n + row; }
  else { row -= B_ * T_; src = tgt + (size_t)row * D_; dst = tgt_n + row; }
  float s = 0.f;
  for (int j = lane; j < D_; j += 32) { float v = src[j]; s += v * v; }
  for (int off = 16; off; off >>= 1) s += __shfl_xor(s, off, 32);
  if (lane == 0) *dst = sqrtf(s);
}

// ---------------------------------------------------------------------------
// Kernel 3: cos[b] = (dec[b] x tgt[b]^T) / (|dec||tgt|) via WMMA.
// One block = (batch, 16-row tile); A tile staged in LDS; B rows of tgt read
// as contiguous float2 (b64) since B = tgt^T puts K contiguous per lane.
// ---------------------------------------------------------------------------
__global__ __launch_bounds__(256) void cos_gemm_kernel(
    const float* __restrict__ dec,   // [B,T,D]
    const float* __restrict__ tgt,   // [B,T,D]
    const float* __restrict__ dec_n, // [B*T]
    const float* __restrict__ tgt_n, // [B*T]
    float* __restrict__ cosm)        // [B,T,T]
{
  __shared__ float ldsA[16][D_ + 2];
  __shared__ float dn[16];
  const int tid = threadIdx.x;
  const int b = blockIdx.y;
  const int m0 = blockIdx.x * 16;

  const float* decB = dec + ((size_t)b * T_ + m0) * D_;
  for (int idx = tid; idx < 16 * D_; idx += 256) {
    int r = idx >> 9, c = idx & (D_ - 1);
    ldsA[r][c] = decB[(size_t)r * D_ + c];
  }
  if (tid < 16) dn[tid] = dec_n[b * T_ + m0 + tid];
  __syncthreads();

  const int lane = tid & 31;
  const int wave = tid >> 5;
  const int row16 = lane & 15;
  const int khalf = (lane >> 4) << 1;
  const int nb = wave * 32;                       // each wave: 2 N-tiles
  const float* tgtB = tgt + (size_t)b * T_ * D_;
  const float* tp0 = tgtB + (size_t)(nb + row16) * D_ + khalf;
  const float* tp1 = tgtB + (size_t)(nb + 16 + row16) * D_ + khalf;

  v8f acc0 = {}, acc1 = {};
  for (int k0 = 0; k0 < D_; k0 += 4) {
    v2f a;
    a.x = ldsA[row16][k0 + khalf];
    a.y = ldsA[row16][k0 + khalf + 1];
    v2f b0; b0.x = tp0[k0]; b0.y = tp0[k0 + 1];   // contiguous -> b64 load
    v2f b1; b1.x = tp1[k0]; b1.y = tp1[k0 + 1];
    acc0 = wmma_f32_16x16x4(a, b0, acc0);
    acc1 = wmma_f32_16x16x4(a, b1, acc1);
  }

  const int rhi = (lane >> 4) * 8;
  const float tn0 = tgt_n[b * T_ + nb + row16];
  const float tn1 = tgt_n[b * T_ + nb + 16 + row16];
  float* cB = cosm + (size_t)b * T_ * T_;
  for (int j = 0; j < 8; ++j) {
    const int rr = rhi + j;
    const float d = dn[rr];
    cB[(size_t)(m0 + rr) * T_ + nb + row16]      = acc0[j] / (d * tn0);
    cB[(size_t)(m0 + rr) * T_ + nb + 16 + row16] = acc1[j] / (d * tn1);
  }
}

// ---------------------------------------------------------------------------
// Kernel 4: greedy assignment. One block per batch; whole 256x256 matrix in
// LDS (row stride 257 -> conflict-free row scans AND column writes, since
// 257 mod 64 == 1). Per-row cached maxima; only rows whose cached argmax
// column is eliminated get rescanned (parallel, 16 threads per row).
// Dynamic LDS: 267,344 B (<= 320KB/WGP).
// ---------------------------------------------------------------------------
#define MAT_STRIDE 257
#define HSP_SMEM ((size_t)(T_ * MAT_STRIDE + 4 * T_ + 8 + 8 + 4) * 4)

__global__ __launch_bounds__(256) void hsp_greedy_kernel(
    const float* __restrict__ cosm,          // [B,T,T]
    const unsigned char* __restrict__ tmask, // [B,T]
    int* __restrict__ posOut)                // [B,T]
{
  extern __shared__ char smem[];
  float* mat   = (float*)smem;                   // T_*MAT_STRIDE
  float* rmv   = mat + T_ * MAT_STRIDE;          // per-row cached max value
  int*   rmc   = (int*)(rmv + T_);               // per-row cached max col
  int*   posl  = rmc + T_;                       // result positions
  int*   wl    = posl + T_;                      // rescan worklist
  float* red_v = (float*)(wl + T_);              // 8 per-wave maxima
  int*   red_i = (int*)(red_v + 8);
  int*   ctrl  = red_i + 8;  // [0]=slen [1]=best_r [2]=best_c [3]=wl_count

  const int tid = threadIdx.x;
  const int b = blockIdx.x;
  const int lane = tid & 31;
  const int wave = tid >> 5;
  const int g = tid >> 4, l16 = tid & 15;

  // slen = popcount(tgt_mask[b]) (prefix mask)
  {
    int m = tmask[b * T_ + tid] ? 1 : 0;
    unsigned long long bl = __ballot(m);
    if (lane == 0) red_i[wave] = __popcll(bl);
  }
  __syncthreads();
  if (tid == 0) {
    int s = 0;
    for (int w = 0; w < 8; ++w) s += red_i[w];
    ctrl[0] = s;
  }
  __syncthreads();
  const int slen = ctrl[0];

  // Load matrix with validity mask (invalid -> -inf; NaN-guard like jnp.where)
  const float* cB = cosm + (size_t)b * T_ * T_;
  for (int idx = tid; idx < T_ * T_; idx += 256) {
    int r = idx >> 8, c = idx & (T_ - 1);
    float v = NEG_INF;
    if (r < slen && c < slen) {
      v = cB[idx];
      if (!(v == v)) v = NEG_INF;
    }
    mat[r * MAT_STRIDE + c] = v;
  }
  __syncthreads();

  // Initial per-row maxima (first occurrence wins ties, like jnp.argmax)
  {
    float bv = NEG_INF;
    int bc = 0;
    const float* row = mat + tid * MAT_STRIDE;
    for (int c = 0; c < T_; ++c) {
      float v = row[c];
      if (v > bv) { bv = v; bc = c; }
    }
    rmv[tid] = bv; rmc[tid] = bc; posl[tid] = 0;
  }
  __syncthreads();

  for (int i = 0; i < slen; ++i) {
    // (1) argmax over row maxima; tie -> smaller row (row-major flat argmax)
    float v = rmv[tid];
    int ridx = tid;
    for (int off = 16; off; off >>= 1) {
      float ov = __shfl_xor(v, off, 32);
      int oi = __shfl_xor(ridx, off, 32);
      if (ov > v || (ov == v && oi < ridx)) { v = ov; ridx = oi; }
    }
    if (lane == 0) { red_v[wave] = v; red_i[wave] = ridx; }
    if (tid == 0) ctrl[3] = 0;
    __syncthreads();
    if (tid == 0) {
      float bvv = red_v[0];
      int br = red_i[0];
      for (int w = 1; w < 8; ++w) {
        float x = red_v[w];
        if (x > bvv || (x == bvv && red_i[w] < br)) { bvv = x; br = red_i[w]; }
      }
      const int bc = rmc[br];
      ctrl[1] = br; ctrl[2] = bc;
      posl[br] = bc;
      rmv[br] = NEG_INF;                  // remove row
    }
    __syncthreads();
    const int bc_ = ctrl[2];

    // (2) eliminate column bc_ (bank-conflict-free: stride 257)
    mat[tid * MAT_STRIDE + bc_] = -2.0f;
    // (3) rows whose cached max sat in bc_ need a rescan
    if (rmc[tid] == bc_ && rmv[tid] > NEG_INF) {
      int slot = atomicAdd(&ctrl[3], 1);
      wl[slot] = tid;
    }
    __syncthreads();

    // (4) parallel rescan: 16 groups x 16 lanes; first-col-wins tie-breaking
    const int nwl = ctrl[3];
    for (int e = g; e < nwl; e += 16) {
      const int rr = wl[e];
      const float* row = mat + rr * MAT_STRIDE + l16 * 16;
      float lbv = NEG_INF;
      int lbc = l16 * 16;
      for (int j = 0; j < 16; ++j) {
        float x = row[j];
        if (x > lbv) { lbv = x; lbc = l16 * 16 + j; }
      }
      for (int off = 8; off; off >>= 1) {
        float ov = __shfl_xor(lbv, off, 16);
        int oc = __shfl_xor(lbc, off, 16);
        if (ov > lbv || (ov == lbv && oc < lbc)) { lbv = ov; lbc = oc; }
      }
      if (l16 == 0) { rmv[rr] = lbv; rmc[rr] = lbc; }
    }
    __syncthreads();
  }

  posOut[b * T_ + tid] = posl[tid];
}

// ---------------------------------------------------------------------------
// Kernel 5: out = dec_input + pos_emb[pos]  (pure bandwidth, b128 traffic)
// ---------------------------------------------------------------------------
__global__ __launch_bounds__(128) void finalize_kernel(
    const float* __restrict__ dec,      // [B,T,D]
    const float* __restrict__ pos_emb,  // [T,D]
    const int* __restrict__ pos,        // [B*T]
    float* __restrict__ out)            // [B,T,D]
{
  const int row = blockIdx.x;           // b*T + t
  const int p = pos[row];
  const float4* a = (const float4*)(dec + (size_t)row * D_);
  const float4* e = (const float4*)(pos_emb + (size_t)p * D_);
  float4* o = (float4*)(out + (size_t)row * D_);
  const int j = threadIdx.x;            // 128 threads x float4 = 512 floats
  float4 x = a[j], y = e[j];
  float4 r;
  r.x = x.x + y.x; r.y = x.y + y.y; r.z = x.z + y.z; r.w = x.w + y.w;
  o[j] = r;
}

// ---------------------------------------------------------------------------
extern "C" void kernel_launch(void* const* d_in, const int* in_sizes, int n_in,
                              void* d_out, int out_size, void* d_ws,
                              size_t ws_size, hipStream_t stream) {
  (void)in_sizes; (void)n_in; (void)out_size; (void)ws_size;
  const float* enc  = (const float*)d_in[0];            // [B,S,D]
  const float* tgt  = (const float*)d_in[1];            // [B,T,D]
  const float* pemb = (const float*)d_in[2];            // [T,D]
  const unsigned char* smask = (const unsigned char*)d_in[3];  // bool bytes
  const unsigned char* tmask = (const unsigned char*)d_in[4];  // bool bytes

  float* outF = (float*)d_out;                          // [B,T,D] f32
  int* outPos = (int*)(outF + (size_t)B_ * T_ * D_);    // [B,T] i32 bits

  float* ws    = (float*)d_ws;
  float* dec   = ws;                                    // B*T*D
  float* cosm  = dec + (size_t)B_ * T_ * D_;            // B*T*T
  float* dec_n = cosm + (size_t)B_ * T_ * T_;           // B*T
  float* tgt_n = dec_n + (size_t)B_ * T_;               // B*T

  soft_copy_kernel<<<dim3(T_ / 16, B_), dim3(256), 0, stream>>>(
      enc, smask, tmask, dec);
  norms_kernel<<<dim3(2 * B_ * T_ / 8), dim3(256), 0, stream>>>(
      dec, tgt, dec_n, tgt_n);
  cos_gemm_kernel<<<dim3(T_ / 16, B_), dim3(256), 0, stream>>>(
      dec, tgt, dec_n, tgt_n, cosm);

  hipFuncSetAttribute((const void*)hsp_greedy_kernel,
                      hipFuncAttributeMaxDynamicSharedMemorySize,
                      (int)HSP_SMEM);
  hsp_greedy_kernel<<<dim3(B_), dim3(256), HSP_SMEM, stream>>>(
      cosm, tmask, outPos);

  finalize_kernel<<<dim3(B_ * T_), dim3(128), 0, stream>>>(
      dec, pemb, outPos, outF);
}